// SimilarityLayer_63969242906928
// MI455X (gfx1250) — compile-verified
//
#include <hip/hip_runtime.h>

typedef __attribute__((ext_vector_type(2))) float v2f;
typedef __attribute__((ext_vector_type(4))) float v4f;
typedef __attribute__((ext_vector_type(8))) float v8f;

#define N_DIM 1024
#define M_DIM 1024
#define T_DIM 64
#define D_DIM 128

#define TN 64    // n rows per block
#define TM 64    // m rows per block
#define TT 4     // consecutive t per block (vectorized output stores)
#define KC 64    // K chunk staged in LDS
#define LDP 68   // padded LDS row length in floats (68 % 64 == 4 -> conflict-free)

// Block: 256 threads = 8 waves. Wave tile: 16n x 32m x 4t (8 accumulators).
// LDS: As[256][68] + Bs[256][68] + sq1[256] + sq2[256] = 141,312 bytes.
// 2 blocks/WGP (282 KB of 320 KB LDS) -> 4 waves/SIMD; cap VGPRs to match.
__global__ __launch_bounds__(256)
__attribute__((amdgpu_waves_per_eu(4)))
void similarity_wmma_kernel(const float* __restrict__ X1,
                            const float* __restrict__ X2,
                            float* __restrict__ Y) {
    extern __shared__ float smem[];
    float* As  = smem;                   // [ (n_local*4 + t_local) ][ LDP ]
    float* Bs  = smem + 256 * LDP;       // [ (m_local*4 + t_local) ][ LDP ]
    float* sq1 = smem + 2 * 256 * LDP;   // [256] sum_d X1^2 per (n_local,t)
    float* sq2 = sq1 + 256;              // [256] sum_d X2^2 per (m_local,t)

    const int tid  = threadIdx.x;
    const int lane = tid & 31;
    const int wid  = tid >> 5;

    const int n0 = blockIdx.z * TN;
    const int m0 = blockIdx.y * TM;
    const int t0 = blockIdx.x * TT;   // t-chunk fastest-varying: L2 write-combining

    // wave tile position inside the 64x64 block tile
    const int wn = (wid & 3) * 16;    // local n base (4 waves across n)
    const int wm = (wid >> 2) * 32;   // local m base (2 waves across m)
    const int h  = lane >> 4;         // half-wave (K pair select / M high bit)
    const int r  = lane & 15;         // row-in-fragment

    v8f acc[2][TT];
#pragma unroll
    for (int mt = 0; mt < 2; ++mt)
#pragma unroll
        for (int t = 0; t < TT; ++t)
            acc[mt][t] = (v8f)0.0f;

#pragma unroll
    for (int c = 0; c < 2; ++c) {
        const int kc = c * KC;

        // ---- Stage A tile global -> LDS (64 KB), accumulate |x1|^2 partials
        float rs1[16];
#pragma unroll
        for (int i = 0; i < 16; ++i) {
            const int f   = i * 256 + tid;
            const int row = f >> 4;          // 0..255  == n_local*4 + t_local
            const int d4  = (f & 15) << 2;   // 0,4,...,60
            const int nl  = row >> 2;
            const int tl  = row & 3;
            const v4f a = *(const v4f*)(X1 + (size_t)(n0 + nl) * (T_DIM * D_DIM)
                                           + (size_t)(t0 + tl) * D_DIM + kc + d4);
            *(v4f*)&As[row * LDP + d4] = a;
            rs1[i] = a.x * a.x + a.y * a.y + a.z * a.z + a.w * a.w;
        }

        // ---- Stage B tile global -> LDS (64 KB), accumulate |x2|^2 partials
        float rs2[16];
#pragma unroll
        for (int i = 0; i < 16; ++i) {
            const int f   = i * 256 + tid;
            const int row = f >> 4;
            const int d4  = (f & 15) << 2;
            const int ml  = row >> 2;
            const int tl  = row & 3;
            const v4f b = *(const v4f*)(X2 + (size_t)(m0 + ml) * (T_DIM * D_DIM)
                                           + (size_t)(t0 + tl) * D_DIM + kc + d4);
            *(v4f*)&Bs[row * LDP + d4] = b;
            rs2[i] = b.x * b.x + b.y * b.y + b.z * b.z + b.w * b.w;
        }

        // ---- Reduce squared-norm partials across each 16-lane group and
        //      accumulate into LDS (owning lane r==0 does a private RMW;
        //      barriers on both sides make this race-free).
#pragma unroll
        for (int i = 0; i < 16; ++i) {
            float a = rs1[i], b = rs2[i];
#pragma unroll
            for (int off = 8; off > 0; off >>= 1) {
                a += __shfl_xor(a, off, 32);
                b += __shfl_xor(b, off, 32);
            }
            if (r == 0) {
                const int row = i * 16 + (tid >> 4);
                if (c == 0) { sq1[row] = a;  sq2[row] = b; }
                else        { sq1[row] += a; sq2[row] += b; }
            }
        }
        __syncthreads();

        // ---- Compute: K-step of 8 -> one b128 LDS load feeds two WMMA ops.
        // d-permutation per lane-half is identical for A and B fragments, so
        // the K reduction covers [k0, k0+8) exactly once.
        for (int k0 = 0; k0 < KC; k0 += 8) {
            v4f af[TT];
#pragma unroll
            for (int t = 0; t < TT; ++t)
                af[t] = *(const v4f*)&As[((wn + r) * TT + t) * LDP + k0 + 4 * h];
#pragma unroll
            for (int mt = 0; mt < 2; ++mt) {
#pragma unroll
                for (int t = 0; t < TT; ++t) {
                    const v4f bf = *(const v4f*)&Bs[((wm + mt * 16 + r) * TT + t) * LDP + k0 + 4 * h];
                    const v2f alo = {af[t].x, af[t].y}, ahi = {af[t].z, af[t].w};
                    const v2f blo = {bf.x, bf.y},       bhi = {bf.z, bf.w};
                    acc[mt][t] = __builtin_amdgcn_wmma_f32_16x16x4_f32(
                        false, alo, false, blo, (short)0, acc[mt][t], false, false);
                    acc[mt][t] = __builtin_amdgcn_wmma_f32_16x16x4_f32(
                        false, ahi, false, bhi, (short)0, acc[mt][t], false, false);
                }
            }
        }
        __syncthreads();
    }

    // ---- Epilogue: Y = (|x1|^2 - 2*cross + |x2|^2) / D, float4 over t
    const float inv_d = 1.0f / (float)D_DIM;
#pragma unroll
    for (int mt = 0; mt < 2; ++mt) {
        const int ml = wm + mt * 16 + r;          // local m (N dim of tile)
        const size_t mg = (size_t)(m0 + ml);
#pragma unroll
        for (int v = 0; v < 8; ++v) {
            const int nl = wn + v + 8 * h;        // local n (M dim of tile)
            const size_t ng = (size_t)(n0 + nl);
            v4f o;
#pragma unroll
            for (int t = 0; t < TT; ++t) {
                const float cval = acc[mt][t][v];
                o[t] = (sq1[nl * TT + t] - 2.0f * cval + sq2[ml * TT + t]) * inv_d;
            }
            *(v4f*)&Y[(ng * M_DIM + mg) * T_DIM + t0] = o;
        }
    }
}

extern "C" void kernel_launch(void* const* d_in, const int* in_sizes, int n_in,
                              void* d_out, int out_size, void* d_ws, size_t ws_size,
                              hipStream_t stream) {
    (void)in_sizes; (void)n_in; (void)out_size; (void)d_ws; (void)ws_size;
    const float* X1 = (const float*)d_in[0];
    const float* X2 = (const float*)d_in[1];
    float* Y = (float*)d_out;

    dim3 grid(T_DIM / TT, M_DIM / TM, N_DIM / TN);   // (16, 16, 16)
    dim3 block(256);
    const size_t shmem = (size_t)(2 * 256 * LDP + 2 * 256) * sizeof(float); // 141,312 B
    similarity_wmma_kernel<<<grid, block, shmem, stream>>>(X1, X2, Y);
}